// RGCNLayer_73942156968057
// MI455X (gfx1250) — compile-verified
//
#include <hip/hip_runtime.h>
#include <hip/hip_bf16.h>

typedef float v2f __attribute__((ext_vector_type(2)));
typedef float v8f __attribute__((ext_vector_type(8)));

#define N_NODES 50000
#define N_EDGES 100000
#define N_RELS  64
#define D       64
#define ROW_TILES (N_NODES / 16)   // 3125, exact

// ---------------- setup kernels ----------------

__global__ void k_init(int* last_edge, int* rel_count, int* cursor) {
  int i = blockIdx.x * blockDim.x + threadIdx.x;
  if (i < N_NODES) last_edge[i] = -1;
  if (i < N_RELS) { rel_count[i] = 0; cursor[i] = 0; }
}

__global__ void k_lastedge(const int* __restrict__ edges, int* __restrict__ last_edge) {
  int e = blockIdx.x * blockDim.x + threadIdx.x;
  if (e < N_EDGES) atomicMax(&last_edge[edges[e * 3 + 2]], e);
}

__global__ void k_count(const int* __restrict__ edges, const int* __restrict__ last_edge,
                        int* __restrict__ rel_count) {
  int v = blockIdx.x * blockDim.x + threadIdx.x;
  if (v < N_NODES) {
    int e = last_edge[v];
    if (e >= 0) atomicAdd(&rel_count[edges[e * 3 + 1]], 1);
  }
}

__global__ void k_prefix(const int* __restrict__ rel_count, int* __restrict__ rel_offset) {
  if (blockIdx.x == 0 && threadIdx.x == 0) {
    int acc = 0;
    for (int r = 0; r < N_RELS; ++r) {
      rel_offset[r] = acc;
      acc += rel_count[r];
    }
    rel_offset[N_RELS] = acc;
  }
}

__global__ void k_scatter(const int* __restrict__ edges, const int* __restrict__ last_edge,
                          const int* __restrict__ rel_offset, int* __restrict__ cursor,
                          int* __restrict__ src_node, int* __restrict__ out_node) {
  int v = blockIdx.x * blockDim.x + threadIdx.x;
  if (v < N_NODES) {
    int e = last_edge[v];
    if (e >= 0) {
      int r = edges[e * 3 + 1];
      int pos = rel_offset[r] + atomicAdd(&cursor[r], 1);
      src_node[pos] = edges[e * 3 + 0];
      out_node[pos] = v;
    }
  }
}

// ---------------- async LDS staging of one 64x64 f32 matrix ----------------
// CDNA5 async global->LDS DMA path: each lane moves 16B; LDS dest address is
// the low 32 bits of the generic shared-memory pointer (HW adds wave LDS_BASE).
__device__ __forceinline__ void stage_matrix_async(float* sW, const float* __restrict__ gW) {
  const int nthreads = blockDim.x;           // 256
  for (int c = threadIdx.x; c < (D * D / 4); c += nthreads) {
    unsigned ldst = (unsigned)(uintptr_t)(sW + c * 4);
    const float* g = gW + c * 4;
    asm volatile("global_load_async_to_lds_b128 %0, %1, off"
                 :: "v"(ldst), "v"(g)
                 : "memory");
  }
  asm volatile("s_wait_asynccnt 0x0" ::: "memory");
}

// ---------------- WMMA GEMM: out = h @ W_self ----------------
// W_self staged once per block into LDS; 8 waves x one 16-row strip each.
__global__ void k_selfloop_gemm(const float* __restrict__ h, const float* __restrict__ W,
                                float* __restrict__ out) {
  __shared__ __align__(16) float sW[D * D];
  stage_matrix_async(sW, W);
  __syncthreads();

  int wave = blockIdx.x * (blockDim.x >> 5) + (threadIdx.x >> 5);
  if (wave >= ROW_TILES) return;            // after barrier; whole-wave uniform
  int lane  = threadIdx.x & 31;
  int m     = lane & 15;                    // A row / B,C,D column within tile
  int khalf = (lane >> 4) << 1;             // 0 or 2 (K pair per half-wave)
  int row0  = wave * 16;

  v8f acc[4] = {};
  const float* arow = h + (size_t)(row0 + m) * D;
  for (int k = 0; k < D; k += 4) {
    v2f a;
    {
      const float2 av = *(const float2*)(arow + k + khalf);   // contiguous K pair
      a.x = av.x; a.y = av.y;
    }
#pragma unroll
    for (int j = 0; j < 4; ++j) {
      v2f b;
      b.x = sW[(k + khalf)     * D + j * 16 + m];
      b.y = sW[(k + khalf + 1) * D + j * 16 + m];
      acc[j] = __builtin_amdgcn_wmma_f32_16x16x4_f32(false, a, false, b,
                                                     (short)0, acc[j], false, false);
    }
  }

  // D layout: lane holds rows (lane>=16 ? 8 : 0)+i, col = lane&15
  int n  = lane & 15;
  int mh = (lane >> 4) << 3;
#pragma unroll
  for (int j = 0; j < 4; ++j)
#pragma unroll
    for (int i = 0; i < 8; ++i)
      out[(size_t)(row0 + mh + i) * D + j * 16 + n] = acc[j][i];
}

// ---------------- WMMA grouped GEMM: one relation per block ----------------
// Block r stages weight[r] into LDS once; its 8 waves sweep that relation's
// 16-row tiles (gather A rows via src_node, scatter += via out_node).
__global__ void k_relagg_gemm(const float* __restrict__ h, const float* __restrict__ weight,
                              const int* __restrict__ src_node, const int* __restrict__ out_node,
                              const int* __restrict__ rel_offset,
                              float* __restrict__ out) {
  __shared__ __align__(16) float sW[D * D];
  int r = blockIdx.x;                                  // 64 blocks, one per relation
  stage_matrix_async(sW, weight + (size_t)r * D * D);
  __syncthreads();

  int begin = rel_offset[r];
  int end   = rel_offset[r + 1];
  int ntile = (end - begin + 15) >> 4;

  int wavein = threadIdx.x >> 5;                       // 0..7
  int nwaves = blockDim.x >> 5;
  int lane   = threadIdx.x & 31;
  int m      = lane & 15;
  int khalf  = (lane >> 4) << 1;
  int n      = lane & 15;
  int mh     = (lane >> 4) << 3;

  for (int t = wavein; t < ntile; t += nwaves) {       // wave-uniform loop
    int row0 = begin + t * 16;

    int gi = row0 + m;
    int s  = (gi < end) ? src_node[gi] : 0;            // pad rows read row 0 (ignored on store)
    const float* arow = h + (size_t)s * D;

    v8f acc[4] = {};
    for (int k = 0; k < D; k += 4) {
      v2f a;
      {
        const float2 av = *(const float2*)(arow + k + khalf);
        a.x = av.x; a.y = av.y;
      }
#pragma unroll
      for (int j = 0; j < 4; ++j) {
        v2f b;
        b.x = sW[(k + khalf)     * D + j * 16 + m];
        b.y = sW[(k + khalf + 1) * D + j * 16 + m];
        acc[j] = __builtin_amdgcn_wmma_f32_16x16x4_f32(false, a, false, b,
                                                       (short)0, acc[j], false, false);
      }
    }

#pragma unroll
    for (int i = 0; i < 8; ++i) {
      int gr = row0 + mh + i;
      if (gr < end) {                                  // divergence only after WMMA
        int v = out_node[gr];
        float* orow = out + (size_t)v * D;
#pragma unroll
        for (int j = 0; j < 4; ++j)
          orow[j * 16 + n] += acc[j][i];               // unique rows per tile: no atomics
      }
    }
  }
}

// ---------------- launch ----------------

extern "C" void kernel_launch(void* const* d_in, const int* in_sizes, int n_in,
                              void* d_out, int out_size, void* d_ws, size_t ws_size,
                              hipStream_t stream) {
  const float* h      = (const float*)d_in[0];
  const int*   edges  = (const int*)d_in[1];
  const float* weight = (const float*)d_in[2];
  const float* wself  = (const float*)d_in[3];
  float*       out    = (float*)d_out;

  char* w = (char*)d_ws;
  int* last_edge  = (int*)w;  w += (size_t)N_NODES * sizeof(int);
  int* rel_count  = (int*)w;  w += (size_t)N_RELS * sizeof(int);
  int* cursor     = (int*)w;  w += (size_t)N_RELS * sizeof(int);
  int* rel_offset = (int*)w;  w += (size_t)(N_RELS + 1) * sizeof(int);
  int* src_node   = (int*)w;  w += (size_t)N_NODES * sizeof(int);
  int* out_node   = (int*)w;  w += (size_t)N_NODES * sizeof(int);

  const int B = 256;
  k_init    <<<(N_NODES + B - 1) / B, B, 0, stream>>>(last_edge, rel_count, cursor);
  k_lastedge<<<(N_EDGES + B - 1) / B, B, 0, stream>>>(edges, last_edge);
  k_count   <<<(N_NODES + B - 1) / B, B, 0, stream>>>(edges, last_edge, rel_count);
  k_prefix  <<<1, 32, 0, stream>>>(rel_count, rel_offset);
  k_scatter <<<(N_NODES + B - 1) / B, B, 0, stream>>>(edges, last_edge, rel_offset, cursor,
                                                      src_node, out_node);

  // 8 waves/block; 3125 row tiles
  k_selfloop_gemm<<<(ROW_TILES + 7) / 8, 256, 0, stream>>>(h, wself, out);

  // one relation per block
  k_relagg_gemm<<<N_RELS, 256, 0, stream>>>(h, weight, src_node, out_node, rel_offset, out);
}